// AttentionReader_62380105007454
// MI455X (gfx1250) — compile-verified
//
#include <hip/hip_runtime.h>
#include <cstdint>

// ---------------- problem constants (from reference setup) ----------------
#define L_CTX     32768
#define DIM       256
#define VOCAB     266
#define MEM_TOK   7
#define BYTE_BASE 10

// tile: 4088 scanned tokens + 8 overlap = 4096 tokens = 16 KB = exactly
// 1024 b128 chunks = 4 async loads per thread (uniform ASYNCcnt per wave)
#define TILE_TOK  4088
#define BUF_TOK   4096
#define NTILES    ((L_CTX + TILE_TOK - 1) / TILE_TOK)   // 9

typedef __attribute__((ext_vector_type(2))) float v2f;
typedef __attribute__((ext_vector_type(8))) float v8f;

__global__ __launch_bounds__(256) void attn_reader_kernel(
    const int* __restrict__ ctx, const float* __restrict__ emb,
    const int* __restrict__ pMarker, const int* __restrict__ pAddr,
    float* __restrict__ out, int B)
{
    __shared__ __align__(16) int sbuf[2][BUF_TOK];   // 32 KB double buffer
    __shared__ int sM, sF, sFound, sVt[4];

    const int b   = blockIdx.x;
    const int tid = threadIdx.x;

    const int marker = pMarker[0];
    const int addr   = pAddr[0];
    const int a0 = BYTE_BASE + ( addr        & 255);
    const int a1 = BYTE_BASE + ((addr >>  8) & 255);
    const int a2 = BYTE_BASE + ((addr >> 16) & 255);
    const int a3 = BYTE_BASE + ((addr >> 24) & 255);

    const int* row = ctx + (size_t)b * L_CTX;
    const unsigned rowBytes = L_CTX * 4u;

    if (tid == 0) { sM = -1; sF = -1; }

    // ---- stage one 16 KB tile (4096 tokens incl. 8-token overlap) into LDS
    //      via async global->LDS b128; addresses clamped (never predicated)
    //      so every wave issues exactly 4 ops -> uniform ASYNCcnt.
    auto stage = [&](int which, int tile) {
        unsigned ldsBase = (unsigned)(uintptr_t)(&sbuf[which][0]);
        unsigned gBase   = (unsigned)tile * (TILE_TOK * 4u);
        #pragma unroll
        for (int j = 0; j < 4; ++j) {
            unsigned c = (unsigned)tid + 256u * (unsigned)j;   // chunk id 0..1023
            unsigned g = gBase + c * 16u;
            if (g > rowBytes - 16u) g = rowBytes - 16u;        // tail clamp (dup loads hit L2)
            unsigned l = ldsBase + c * 16u;
            asm volatile("global_load_async_to_lds_b128 %0, %1, %2"
                         :: "v"(l), "v"(g), "s"(row) : "memory");
        }
    };

    int mMax = -1, fMax = -1;

    stage(0, 0);
    for (int t = 0; t < NTILES; ++t) {
        if (t + 1 < NTILES) {
            stage((t + 1) & 1, t + 1);                         // prefetch next tile
            asm volatile("s_wait_asynccnt 4" ::: "memory");    // drain tile t only
        } else {
            asm volatile("s_wait_asynccnt 0" ::: "memory");
        }
        __syncthreads();

        const int cur   = t & 1;
        const int t0    = t * TILE_TOK;
        const int valid = min(TILE_TOK, L_CTX - t0);
        const int4* b4  = (const int4*)(&sbuf[cur][0]);

        #pragma unroll
        for (int j = 0; j < 4; ++j) {
            int  c = tid + 256 * j;
            int4 v = b4[c];                                    // ds_load_b128
            int tk[4] = { v.x, v.y, v.z, v.w };
            #pragma unroll
            for (int e = 0; e < 4; ++e) {
                int lp = 4 * c + e;
                int p  = t0 + lp;
                bool in = lp < valid;
                if (in && tk[e] == marker) mMax = max(mMax, p);
                if (in && tk[e] == MEM_TOK && (p + 8 < L_CTX)) {
                    // rare path (once per row): overlap region covers lp+4
                    if (sbuf[cur][lp + 1] == a0 && sbuf[cur][lp + 2] == a1 &&
                        sbuf[cur][lp + 3] == a2 && sbuf[cur][lp + 4] == a3)
                        fMax = max(fMax, p);
                }
            }
        }
        __syncthreads();
    }

    atomicMax(&sM, mMax);
    atomicMax(&sF, fMax);
    __syncthreads();

    // ---- decode register / memory values (thread 0) ----
    if (tid == 0) {
        int mpos = sM, fpos = sF;
        unsigned reg = 0;
        if (mpos >= 0) {
            for (int j = 0; j < 4; ++j) {
                int idx = min(max(mpos + 1 + j, 0), L_CTX - 1);
                int tkn = row[idx];
                unsigned by = (unsigned)min(max(tkn - BYTE_BASE, 0), 255);
                reg |= by << (8 * j);
            }
        }
        int found = (fpos >= 0) ? 1 : 0;
        unsigned mv = 0;
        for (int j = 0; j < 4; ++j) {
            int idx = min(max(fpos + 5 + j, 0), L_CTX - 1);
            int tkn = row[idx];
            unsigned by = (unsigned)min(max(tkn - BYTE_BASE, 0), 255);
            mv |= by << (8 * j);
            sVt[j] = min(max(tkn, 0), VOCAB - 1);              // emb row index (clipped)
        }
        if (!found) mv = 0;
        sFound = found;
        out[b]     = (float)reg;
        out[B + b] = (float)mv;
    }
    __syncthreads();

    // ---- embedding gather as exact one-hot WMMA: D = I(4x4 in 16x4) * Brows
    //      f32 WMMA: products are 1.0*x or 0.0*x -> bit-exact gather.
    if (tid < 32) {                                            // wave 0, EXEC all ones
        const int lane = tid;
        const int m    = lane & 15;
        const int hi   = lane >> 4;                            // 0 -> K {0,1}, 1 -> K {2,3}
        v2f a;
        a.x = (m == 2 * hi + 0) ? 1.0f : 0.0f;                 // A[m, K=2*hi]
        a.y = (m == 2 * hi + 1) ? 1.0f : 0.0f;                 // A[m, K=2*hi+1]
        const int n  = lane & 15;
        const int r0 = sVt[2 * hi + 0];
        const int r1 = sVt[2 * hi + 1];
        const bool fnd = (sFound != 0);
        float* ov = out + 2 * B + (size_t)b * (4 * DIM);
        for (int tc = 0; tc < 16; ++tc) {
            int col = tc * 16 + n;
            v2f bm;
            bm.x = fnd ? emb[r0 * DIM + col] : 0.0f;           // B row K=2*hi
            bm.y = fnd ? emb[r1 * DIM + col] : 0.0f;           // B row K=2*hi+1
            v8f cc = {0.f,0.f,0.f,0.f,0.f,0.f,0.f,0.f};
            v8f d = __builtin_amdgcn_wmma_f32_16x16x4_f32(
                        false, a, false, bm, (short)0, cc, false, false);
            if (lane < 16) {                                   // rows m=0..3 live in d[0..3]
                ov[0 * DIM + col] = d[0];
                ov[1 * DIM + col] = d[1];
                ov[2 * DIM + col] = d[2];
                ov[3 * DIM + col] = d[3];
            }
        }
    }
}

extern "C" void kernel_launch(void* const* d_in, const int* in_sizes, int n_in,
                              void* d_out, int out_size, void* d_ws, size_t ws_size,
                              hipStream_t stream) {
    const int*   ctx = (const int*)d_in[0];
    const float* emb = (const float*)d_in[1];
    const int*   pm  = (const int*)d_in[2];
    const int*   pa  = (const int*)d_in[3];
    float*       out = (float*)d_out;
    const int B = in_sizes[0] / L_CTX;                         // 512
    attn_reader_kernel<<<B, 256, 0, stream>>>(ctx, emb, pm, pa, out, B);
}